// ARIMAModel_20255065768547
// MI455X (gfx1250) — compile-verified
//
#include <hip/hip_runtime.h>
#include <hip/hip_bf16.h>

typedef __attribute__((ext_vector_type(2))) float v2f;
typedef __attribute__((ext_vector_type(8))) float v8f;

#define BSZ    8192
#define TSEQ   4096
#define OUTW   4095   // (TSEQ - D - P) + Q = 4093 + 2
#define NVALID 4093   // valid eps indices k = 0..4092
#define CH     16     // WMMA time-chunk
#define NCHUNK 256    // 256*16 = 4096 covers all k (tail guarded)
#define WAVES_PER_BLOCK 2
#define ROWS_PER_WAVE   16
#define ROWS_PER_BLOCK  (WAVES_PER_BLOCK * ROWS_PER_WAVE)
#define PF_DIST (8 * CH)   // prefetch 8 chunks (~512B) ahead per row

// One 16-time-step chunk for 16 rows handled by one wave.
// eps(chunk) = H_toeplitz @ targets  +  e1*h[m+1] + e2*(-th1*h[m])
template <bool LAST>
__device__ __forceinline__ void process_chunk(
    int base, const float* __restrict__ yrow, float* __restrict__ orow,
    int khalf, int m0, int lrow,
    const float A0k[4], const float A1k[4], const float hv[9],
    float q0, float ph1, float muv, float th1,
    float& e1, float& e2)
{
  // Private sliding y window: yb[j] = y[row, base + khalf + j], j = 1..16.
  float yb[17];
  const int tbase = base + khalf;
#pragma unroll
  for (int j = 1; j <= 16; ++j) {
    if (LAST) {
      const int idx = tbase + j;
      yb[j] = (idx < TSEQ) ? yrow[idx] : 0.0f;
    } else {
      yb[j] = yrow[tbase + j];
    }
  }
  if (!LAST) {
    // GL2 prefetch 512B ahead of the streaming loads. locality 0 -> SCOPE_SYS,
    // which is a *speculative* prefetch (OOB silently dropped) and folds into
    // the load address register + immediate offset at zero VALU cost.
    // (WGP-scope prefetch is not reachable: LLVM clamps locality 3 to SE,
    //  and SE/DEV/SYS all fill GL2 only.)
    __builtin_prefetch(yrow + tbase + PF_DIST, 0, 0);
  }

  // Zero-state response via 4 accumulated f32 WMMAs (K = 4 each).
  v8f acc = {0.f, 0.f, 0.f, 0.f, 0.f, 0.f, 0.f, 0.f};
#pragma unroll
  for (int kk = 0; kk < 4; ++kk) {
    const int o = 4 * kk;
    // target(k) = y[k+3] - (1+ph0)*y[k+2] - ph1*y[k+1] - mu, k = base+khalf+4kk+c
    const float t0 = fmaf(-ph1, yb[o + 1], fmaf(-q0, yb[o + 2], yb[o + 3] - muv));
    const float t1 = fmaf(-ph1, yb[o + 2], fmaf(-q0, yb[o + 3], yb[o + 4] - muv));
    v2f a; a.x = A0k[kk]; a.y = A1k[kk];   // A = H[:, 4kk : 4kk+4]
    v2f b; b.x = t0;      b.y = t1;        // B = targets (4 x 16 rows)
    acc = __builtin_amdgcn_wmma_f32_16x16x4_f32(
        false, a, false, b, (short)0, acc, false, false);
  }

  // Homogeneous carry fix-up (exact): d[m] = acc[m] + e1*h[m+1] - th1*e2*h[m]
  const float e2m = -th1 * e2;
  float d[8];
#pragma unroll
  for (int v = 0; v < 8; ++v)
    d[v] = fmaf(e1, hv[v + 1], fmaf(e2m, hv[v], acc[v]));

  // New carries = eps at chunk rows m=15 (VGPR7) and m=14 (VGPR6), lanes 16..31.
  e1 = __shfl(d[7], lrow + 16, 32);
  e2 = __shfl(d[6], lrow + 16, 32);

  // Each lane stores 8 contiguous outputs for its row.
  float* op = orow + base + m0;
#pragma unroll
  for (int v = 0; v < 8; ++v) {
    if (LAST) {
      const int k = base + m0 + v;
      if (k < NVALID)     op[v] = d[v];
      else if (k < OUTW)  op[v] = 0.0f;   // trailing Q zeros
    } else {
      op[v] = d[v];
    }
  }
}

__global__ __launch_bounds__(WAVES_PER_BLOCK * 32) void arima_eps_kernel(
    const float* __restrict__ y, const float* __restrict__ phi,
    const float* __restrict__ theta, const float* __restrict__ mu,
    float* __restrict__ out)
{
  __shared__ float h_sh[17];   // impulse response h[0..16]

  const float ph0 = phi[0], ph1 = phi[1];
  const float th0 = theta[0], th1 = theta[1];
  const float muv = mu[0];

  if (threadIdx.x == 0) {
    h_sh[0] = 1.0f;
    h_sh[1] = -th0;
    for (int k = 2; k < 17; ++k)
      h_sh[k] = fmaf(-th0, h_sh[k - 1], -th1 * h_sh[k - 2]);
  }
  __syncthreads();

  const int lane  = threadIdx.x & 31;
  const int wid   = threadIdx.x >> 5;
  const int lrow  = lane & 15;       // row within wave tile / M row for A
  const int half  = lane >> 4;
  const int khalf = half * 2;        // K columns 0,1 (lanes<16) or 2,3 (lanes>=16)
  const int m0    = half * 8;        // D rows m0..m0+7 live in this lane's VGPRs
  const int row   = blockIdx.x * ROWS_PER_BLOCK + wid * ROWS_PER_WAVE + lrow;

  const float* yrow = y   + (size_t)row * TSEQ;
  float*       orow = out + (size_t)row * OUTW;

  // Loop-invariant A-matrix entries: H[m][c] = (m>=c) ? h[m-c] : 0
  float A0k[4], A1k[4];
#pragma unroll
  for (int kk = 0; kk < 4; ++kk) {
    const int c0 = 4 * kk + khalf;
    A0k[kk] = (lrow >= c0)     ? h_sh[lrow - c0]     : 0.0f;
    A1k[kk] = (lrow >= c0 + 1) ? h_sh[lrow - c0 - 1] : 0.0f;
  }
  // Loop-invariant fix-up coefficients h[m0..m0+8]
  float hv[9];
#pragma unroll
  for (int v = 0; v < 9; ++v) hv[v] = h_sh[m0 + v];

  const float q0 = 1.0f + ph0;
  float e1 = 0.0f, e2 = 0.0f;   // eps[-1], eps[-2]

  for (int ch = 0; ch < NCHUNK - 1; ++ch)
    process_chunk<false>(ch * CH, yrow, orow, khalf, m0, lrow,
                         A0k, A1k, hv, q0, ph1, muv, th1, e1, e2);
  process_chunk<true>((NCHUNK - 1) * CH, yrow, orow, khalf, m0, lrow,
                      A0k, A1k, hv, q0, ph1, muv, th1, e1, e2);
}

extern "C" void kernel_launch(void* const* d_in, const int* in_sizes, int n_in,
                              void* d_out, int out_size, void* d_ws, size_t ws_size,
                              hipStream_t stream) {
  const float* y     = (const float*)d_in[0];
  const float* phi   = (const float*)d_in[1];
  const float* theta = (const float*)d_in[2];
  const float* mu    = (const float*)d_in[3];
  float* out = (float*)d_out;

  dim3 grid(BSZ / ROWS_PER_BLOCK);     // 256 blocks -> spread across WGPs
  dim3 block(WAVES_PER_BLOCK * 32);    // 2 waves * 16 rows = 32 rows per block
  hipLaunchKernelGGL(arima_eps_kernel, grid, block, 0, stream,
                     y, phi, theta, mu, out);
}